// AffineAugmentationLayer_28681791603204
// MI455X (gfx1250) — compile-verified
//
#include <hip/hip_runtime.h>
#include <hip/hip_bf16.h>

typedef __attribute__((ext_vector_type(2))) float v2f;
typedef __attribute__((ext_vector_type(8))) float v8f;

// ---------------------------------------------------------------------------
// Stage 1: build per-batch affine transforms with WMMA.
// AtA = 8*I exactly (full-factorial corner matrix), so
//   transform = (1/8) * Bmat^T (4x8) x A (8x4)         per batch.
// Pack 4 batches block-diagonally into one 16x16x4 WMMA pair (K=8 -> 2 WMMAs).
// Wave w handles batches 4w..4w+3. Launch: 1 block, 64 threads (2 waves).
// Output: T[b][i][j], i=0..2 rows, j=0..3 cols -> 12 floats per batch.
// ---------------------------------------------------------------------------
__global__ __launch_bounds__(64) void build_transforms_kernel(
    const float* __restrict__ u,   // [8][8][3] uniform(0,1)
    float* __restrict__ T)         // [8][12]
{
  const int lane = threadIdx.x & 31;
  const int wave = threadIdx.x >> 5;
  const int b0   = wave << 2;        // first batch of this wave

  // f32 A-operand layout (16x4, 2 VGPRs):
  //   lanes 0-15 : row M = lane,      vgpr0=K0, vgpr1=K1
  //   lanes 16-31: row M = lane-16,   vgpr0=K2, vgpr1=K3
  const int rowM = lane & 15;
  const int khi  = lane >> 4;        // 0 -> K{0,1}, 1 -> K{2,3}
  const int p    = rowM >> 2;        // batch-in-wave (block index)
  const int i    = rowM & 3;         // Bmat column == transform row
  const int b    = b0 + p;

  v8f acc = {};
#pragma unroll
  for (int chunk = 0; chunk < 2; ++chunk) {   // K = 8 contraction, 2 x K=4
    v2f av, bv;
#pragma unroll
    for (int t = 0; t < 2; ++t) {
      const int k = (khi << 1) | t;
      const int n = (chunk << 2) | k;         // corner index 0..7
      // ---- A-operand: Aop[4p+i][k] = Bmat[b][n][i]
      float aval;
      if (i < 3) {
        const float ci = ((n >> (2 - i)) & 1) ? 1.0f : -1.0f;  // corner coord
        const float uu = u[(b * 8 + n) * 3 + i];
        aval = ci * (0.8f + 0.2f * uu);       // (1-S) + S*u, S = 0.2
      } else {
        aval = 1.0f;                          // homogeneous column
      }
      av[t] = aval;
      // ---- B-operand (mirrored layout: col = lane&15, same K pairs):
      //      Bop[k][4p+j] = A[n][j], here j == i (same low bits of lane)
      float bval;
      if (i < 3) bval = ((n >> (2 - i)) & 1) ? 1.0f : -1.0f;
      else       bval = 1.0f;
      bv[t] = bval;
    }
    acc = __builtin_amdgcn_wmma_f32_16x16x4_f32(
        /*neg_a=*/false, av, /*neg_b=*/false, bv,
        /*c_mod=*/(short)0, acc, /*reuse_a=*/false, /*reuse_b=*/false);
  }

  // C/D layout: vgpr r, lanes 0-15 -> (M=r, N=lane); lanes 16-31 -> (M=r+8, N=lane-16)
#pragma unroll
  for (int r = 0; r < 8; ++r) {
    const int M  = r + ((lane >= 16) ? 8 : 0);
    const int N  = lane & 15;
    const int pp = M >> 2;
    const int ii = M & 3;
    const int jj = N & 3;
    if (((N >> 2) == pp) && (ii < 3)) {
      T[(b0 + pp) * 12 + ii * 4 + jj] = acc[r] * 0.125f;  // /8 from AtA = 8I
    }
  }
}

// ---------------------------------------------------------------------------
// Stage 2: trilinear resample with zero boundary.
// pixel_i = (d-63.5)*Ti0 + (h-63.5)*Ti1 + (w-63.5)*Ti2 + 63.5*(Ti3+1)
// (the normalization factors cancel exactly). One thread = 4 voxels along w.
// Validity is folded into the six 1-D weights -> zero-boundary for free.
// ---------------------------------------------------------------------------
__global__ __launch_bounds__(256) void resample_kernel(
    const float* __restrict__ img,   // [8][128][128][128]
    const float* __restrict__ T,     // [8][12]
    float* __restrict__ out)         // [8][128][128][128]
{
  const int tid = blockIdx.x * 256 + threadIdx.x;   // 4,194,304 threads total
  const int w0  = (tid & 31) << 2;                  // 32 groups of 4 along w
  const int h   = (tid >> 5) & 127;
  const int d   = (tid >> 12) & 127;
  const int b   = tid >> 19;

  // b is uniform per wave (per-batch thread count is a multiple of 32):
  // force scalar loads of the 12 transform coefficients.
  const int bu = __builtin_amdgcn_readfirstlane(b);
  const float* __restrict__ Tb = T + bu * 12;
  const float T00 = Tb[0], T01 = Tb[1], T02 = Tb[2],  T03 = Tb[3];
  const float T10 = Tb[4], T11 = Tb[5], T12 = Tb[6],  T13 = Tb[7];
  const float T20 = Tb[8], T21 = Tb[9], T22 = Tb[10], T23 = Tb[11];

  const float fd = (float)d  - 63.5f;
  const float fh = (float)h  - 63.5f;
  const float fw = (float)w0 - 63.5f;

  float p0 = fd * T00 + fh * T01 + fw * T02 + 63.5f * (T03 + 1.0f);
  float p1 = fd * T10 + fh * T11 + fw * T12 + 63.5f * (T13 + 1.0f);
  float p2 = fd * T20 + fh * T21 + fw * T22 + 63.5f * (T23 + 1.0f);

  const float* __restrict__ imgb = img + ((size_t)bu << 21);  // b * 128^3

  float rv[4];
#pragma unroll
  for (int t = 0; t < 4; ++t) {
    const float q0 = p0 + (float)t * T02;
    const float q1 = p1 + (float)t * T12;
    const float q2 = p2 + (float)t * T22;

    const float f0 = floorf(q0), f1 = floorf(q1), f2 = floorf(q2);
    const int iz = (int)f0, iy = (int)f1, ix = (int)f2;
    const float rz = q0 - f0, ry = q1 - f1, rx = q2 - f2;

    // 1-D weights with zero-boundary validity folded in
    const float wz0 = ((unsigned)iz       < 128u) ? (1.0f - rz) : 0.0f;
    const float wz1 = ((unsigned)(iz + 1) < 128u) ? rz          : 0.0f;
    const float wy0 = ((unsigned)iy       < 128u) ? (1.0f - ry) : 0.0f;
    const float wy1 = ((unsigned)(iy + 1) < 128u) ? ry          : 0.0f;
    const float wx0 = ((unsigned)ix       < 128u) ? (1.0f - rx) : 0.0f;
    const float wx1 = ((unsigned)(ix + 1) < 128u) ? rx          : 0.0f;

    // clamped addresses (invalid corners have zero weight anyway) -> v_med3
    const int zc0 = min(max(iz,     0), 127), zc1 = min(max(iz + 1, 0), 127);
    const int yc0 = min(max(iy,     0), 127), yc1 = min(max(iy + 1, 0), 127);
    const int xc0 = min(max(ix,     0), 127), xc1 = min(max(ix + 1, 0), 127);

    const int rb00 = (zc0 << 14) + (yc0 << 7);
    const int rb01 = (zc0 << 14) + (yc1 << 7);
    const int rb10 = (zc1 << 14) + (yc0 << 7);
    const int rb11 = (zc1 << 14) + (yc1 << 7);

    const float v000 = imgb[rb00 + xc0], v001 = imgb[rb00 + xc1];
    const float v010 = imgb[rb01 + xc0], v011 = imgb[rb01 + xc1];
    const float v100 = imgb[rb10 + xc0], v101 = imgb[rb10 + xc1];
    const float v110 = imgb[rb11 + xc0], v111 = imgb[rb11 + xc1];

    const float w00 = wz0 * wy0, w01 = wz0 * wy1;
    const float w10 = wz1 * wy0, w11 = wz1 * wy1;

    float acc;
    acc = v000 * (w00 * wx0);
    acc = fmaf(v001, w00 * wx1, acc);
    acc = fmaf(v010, w01 * wx0, acc);
    acc = fmaf(v011, w01 * wx1, acc);
    acc = fmaf(v100, w10 * wx0, acc);
    acc = fmaf(v101, w10 * wx1, acc);
    acc = fmaf(v110, w11 * wx0, acc);
    acc = fmaf(v111, w11 * wx1, acc);
    rv[t] = acc;
  }

  // out flat index == tid*4 (+t); w0 is 16B aligned -> single b128 store
  float4 o;
  o.x = rv[0]; o.y = rv[1]; o.z = rv[2]; o.w = rv[3];
  reinterpret_cast<float4*>(out)[tid] = o;
}

extern "C" void kernel_launch(void* const* d_in, const int* in_sizes, int n_in,
                              void* d_out, int out_size, void* d_ws, size_t ws_size,
                              hipStream_t stream) {
  const float* img = (const float*)d_in[0];   // [8,128,128,128,1] f32
  const float* u   = (const float*)d_in[1];   // [8,8,3] f32
  float* out = (float*)d_out;                 // [8,128,128,128,1] f32
  float* T   = (float*)d_ws;                  // [8][12] f32 scratch (fully overwritten)

  build_transforms_kernel<<<1, 64, 0, stream>>>(u, T);

  const int total_threads = 8 * 128 * 128 * (128 / 4);  // 4,194,304
  resample_kernel<<<total_threads / 256, 256, 0, stream>>>(img, T, out);
}